// Scalar_40114994544662
// MI455X (gfx1250) — compile-verified
//
#include <hip/hip_runtime.h>

// ---------------- static config (mirrors reference) ----------------
#define NZP 304
#define NXP 304
#define FLATSZ (NZP * NXP)
#define NSHOTS 8
#define NT 50
#define NREC 300
#define DT 0.001f
#define INV_DX 0.25f       // 1/4.0
#define INV_DX2 0.0625f    // 1/16.0

typedef float v2f __attribute__((ext_vector_type(2)));
typedef float v8f __attribute__((ext_vector_type(8)));

// 4th-order central difference coefficients (scaled by 1/dx, 1/dx^2).
// Branchless: chained selects lower to v_cmp/v_cndmask, no EXEC-mask branches.
__device__ __forceinline__ float c1co(int d) {
    // c1(+-1) = +-2/3/dx, c1(+-2) = -+1/12/dx (antisymmetric)
    const int ad = (d < 0) ? -d : d;
    float b = (ad == 1) ? (2.f / 3.f) * INV_DX
            : ((ad == 2) ? (-1.f / 12.f) * INV_DX : 0.f);
    return (d < 0) ? -b : b;
}
__device__ __forceinline__ float c2co(int d) {
    // c2(0) = -2.5/dx2, c2(+-1) = 4/3/dx2, c2(+-2) = -1/12/dx2 (symmetric)
    const int ad = (d < 0) ? -d : d;
    return (ad == 0) ? -2.5f * INV_DX2
         : ((ad == 1) ? (4.f / 3.f) * INV_DX2
         : ((ad == 2) ? (-1.f / 12.f) * INV_DX2 : 0.f));
}

// One FDTD+PML time step. One wave32 per 16x16 tile; derivatives via
// V_WMMA_F32_16X16X4_F32 as banded 16x20 x 20x16 matmuls (5 K-chunks).
// u_prev is updated in place to become wfc_new (wfp only read at own point).
// zeta updated in place (own-point only). psi double-buffered (neighbor reads).
__global__ __launch_bounds__(32) void fdtd_step(
    const float* __restrict__ u_cur,
    float* __restrict__ u_prev,               // in: wfp, out: wfc_new
    const float* __restrict__ psi_z_old, const float* __restrict__ psi_x_old,
    float* __restrict__ psi_z_new, float* __restrict__ psi_x_new,
    float* __restrict__ zeta_z, float* __restrict__ zeta_x,
    const float* __restrict__ v,
    const float* __restrict__ a_z, const float* __restrict__ b_z, const float* __restrict__ dbdx_z,
    const float* __restrict__ a_x, const float* __restrict__ b_x, const float* __restrict__ dbdx_x)
{
    __shared__ float sU[20][24];    // wavefield halo tile  rows z0-2..z0+17, cols x0-2..x0+17
    __shared__ float sPX[16][24];   // a_x * psi_x          rows z0..z0+15,   cols x0-2..x0+17
    __shared__ float sPZ[20][16];   // a_z * psi_z          rows z0-2..z0+17, cols x0..x0+15

    const int lane = threadIdx.x;
    const int z0 = blockIdx.x * 16;
    const int x0 = blockIdx.y * 16;
    const size_t base = (size_t)blockIdx.z * FLATSZ;

    for (int i = lane; i < 400; i += 32) {
        int r = i / 20, c = i % 20;
        int gz = z0 - 2 + r, gx = x0 - 2 + c;
        float val = 0.f;
        if (gz >= 0 && gz < NZP && gx >= 0 && gx < NXP)
            val = u_cur[base + (size_t)gz * NXP + gx];
        sU[r][c] = val;
    }
    for (int i = lane; i < 320; i += 32) {
        int r = i / 20, c = i % 20;
        int gz = z0 + r, gx = x0 - 2 + c;
        float val = 0.f;
        if (gx >= 0 && gx < NXP)
            val = a_x[gx] * psi_x_old[base + (size_t)gz * NXP + gx];
        sPX[r][c] = val;
    }
    for (int i = lane; i < 320; i += 32) {
        int r = i / 16, c = i % 16;
        int gz = z0 - 2 + r, gx = x0 + c;
        float val = 0.f;
        if (gz >= 0 && gz < NZP)
            val = a_z[gz] * psi_z_old[base + (size_t)gz * NXP + gx];
        sPZ[r][c] = val;
    }
    __syncthreads();

    const int half = lane >> 4;     // 0: lanes 0-15, 1: lanes 16-31
    const int l15  = lane & 15;

    v8f dudx = {}, d2udx = {}, dudz = {}, d2udz = {}, dpsix = {}, dpsiz = {};

    // 16x20 x 20x16 banded products, K in 5 chunks of 4.
    // A-layout (16x4 f32): lane holds M=l15; V0 <-> K=k0+2*half, V1 <-> K+1.
    // B-layout (4x16 f32): lane holds N=l15; low lanes rows {k0,k0+1}, high lanes {k0+2,k0+3}.
    // Banded coeff value c(K - l15 - 2) is identical for the x-stencil B operand
    // and the z-stencil A operand under these layouts, so one pair serves both.
#pragma unroll
    for (int ck = 0; ck < 5; ++ck) {
        const int k0 = ck * 4 + half * 2;
        const int j  = k0 - l15 - 2;           // banded offset for component .x
        v2f ax, apx, bz, bpz, s1, s2;
        ax.x  = sU[l15 + 2][k0];     ax.y  = sU[l15 + 2][k0 + 1];
        apx.x = sPX[l15][k0];        apx.y = sPX[l15][k0 + 1];
        bz.x  = sU[k0][l15 + 2];     bz.y  = sU[k0 + 1][l15 + 2];
        bpz.x = sPZ[k0][l15];        bpz.y = sPZ[k0 + 1][l15];
        s1.x  = c1co(j);             s1.y  = c1co(j + 1);
        s2.x  = c2co(j);             s2.y  = c2co(j + 1);

        dudx  = __builtin_amdgcn_wmma_f32_16x16x4_f32(false, ax,  false, s1,  (short)0, dudx,  false, false);
        d2udx = __builtin_amdgcn_wmma_f32_16x16x4_f32(false, ax,  false, s2,  (short)0, d2udx, false, false);
        dpsix = __builtin_amdgcn_wmma_f32_16x16x4_f32(false, apx, false, s1,  (short)0, dpsix, false, false);
        dudz  = __builtin_amdgcn_wmma_f32_16x16x4_f32(false, s1,  false, bz,  (short)0, dudz,  false, false);
        d2udz = __builtin_amdgcn_wmma_f32_16x16x4_f32(false, s2,  false, bz,  (short)0, d2udz, false, false);
        dpsiz = __builtin_amdgcn_wmma_f32_16x16x4_f32(false, s1,  false, bpz, (short)0, dpsiz, false, false);
    }

    // C/D layout: lane owns column N=l15, rows M = vr + 8*half for vr=0..7.
    const float dt2 = DT * DT;
#pragma unroll
    for (int vr = 0; vr < 8; ++vr) {
        const int m  = vr + half * 8;
        const int gz = z0 + m, gx = x0 + l15;
        const size_t idx = base + (size_t)gz * NXP + gx;

        const float az  = a_z[gz], bzv = b_z[gz], dbz = dbdx_z[gz];
        const float axv = a_x[gx], bxv = b_x[gx], dbx = dbdx_x[gx];
        const float wfc = sU[m + 2][l15 + 2];
        const float wfp = u_prev[idx];
        const float pz  = psi_z_old[idx], px = psi_x_old[idx];
        const float zz  = zeta_z[idx],    zx = zeta_x[idx];
        const float vv  = v[gz * NXP + gx];

        const float tz   = (1.f + bzv) * d2udz[vr] + dbz * dudz[vr] + dpsiz[vr];
        const float tx   = (1.f + bxv) * d2udx[vr] + dbx * dudx[vr] + dpsix[vr];
        const float wsum = (1.f + bzv) * tz + az * zz + (1.f + bxv) * tx + axv * zx;

        u_prev[idx]    = (vv * vv * dt2) * wsum + 2.f * wfc - wfp;   // wfc_new
        psi_z_new[idx] = bzv * dudz[vr] + az * pz;
        psi_x_new[idx] = bxv * dudx[vr] + axv * px;
        zeta_z[idx]    = bzv * tz + az * zz;
        zeta_x[idx]    = bxv * tx + axv * zx;
    }
}

__global__ void gather_rec(const float* __restrict__ u_cur,
                           const int* __restrict__ rec_i,
                           float* __restrict__ out_t)
{
    int i = blockIdx.x * blockDim.x + threadIdx.x;
    if (i >= NSHOTS * NREC) return;
    int shot = i / NREC;
    out_t[i] = u_cur[(size_t)shot * FLATSZ + rec_i[i]];
}

__global__ void inject_src(float* __restrict__ u_new,
                           const int* __restrict__ src_i,
                           const float* __restrict__ amp_t)
{
    int shot = threadIdx.x;
    if (shot < NSHOTS) {
        size_t idx = (size_t)shot * FLATSZ + src_i[shot];
        u_new[idx] += amp_t[shot];
    }
}

__global__ void zero_ws(float* __restrict__ p, size_t n)
{
    size_t i = (size_t)blockIdx.x * blockDim.x + threadIdx.x;
    if (i < n) p[i] = 0.f;
}

extern "C" void kernel_launch(void* const* d_in, const int* in_sizes, int n_in,
                              void* d_out, int out_size, void* d_ws, size_t ws_size,
                              hipStream_t stream)
{
    const float* v      = (const float*)d_in[0];
    const float* amp    = (const float*)d_in[1];   // (NT, NSHOTS, 1)
    const float* a_z    = (const float*)d_in[2];
    const float* b_z    = (const float*)d_in[3];
    const float* dbdx_z = (const float*)d_in[4];
    const float* a_x    = (const float*)d_in[5];
    const float* b_x    = (const float*)d_in[6];
    const float* dbdx_x = (const float*)d_in[7];
    const int* src_i    = (const int*)d_in[8];     // (NSHOTS, 1)
    const int* rec_i    = (const int*)d_in[9];     // (NSHOTS, NREC)
    float* out          = (float*)d_out;           // (NT, NSHOTS, NREC)

    const size_t S = (size_t)NSHOTS * FLATSZ;
    float* ws  = (float*)d_ws;
    float* uA  = ws + 0 * S;                       // wfp / wfc_new (in place)
    float* uB  = ws + 1 * S;                       // wfc
    float* pzA = ws + 2 * S, *pzB = ws + 3 * S;    // psi_z ping-pong
    float* pxA = ws + 4 * S, *pxB = ws + 5 * S;    // psi_x ping-pong
    float* zz  = ws + 6 * S, *zx  = ws + 7 * S;    // zeta (in place)

    {   // zero all state each call (harness does not re-zero between replays)
        size_t n = 8 * S;
        zero_ws<<<(int)((n + 255) / 256), 256, 0, stream>>>(ws, n);
    }

    float* u_cur  = uB;  float* u_prev = uA;
    float* pz_old = pzA; float* pz_new = pzB;
    float* px_old = pxA; float* px_new = pxB;

    const dim3 sgrid(NZP / 16, NXP / 16, NSHOTS);  // 19 x 19 x 8 tiles
    for (int t = 0; t < NT; ++t) {
        gather_rec<<<(NSHOTS * NREC + 255) / 256, 256, 0, stream>>>(
            u_cur, rec_i, out + (size_t)t * NSHOTS * NREC);

        fdtd_step<<<sgrid, 32, 0, stream>>>(
            u_cur, u_prev, pz_old, px_old, pz_new, px_new, zz, zx,
            v, a_z, b_z, dbdx_z, a_x, b_x, dbdx_x);

        inject_src<<<1, 32, 0, stream>>>(u_prev, src_i, amp + (size_t)t * NSHOTS);

        float* tmp = u_cur;  u_cur  = u_prev; u_prev = tmp;
        tmp = pz_old; pz_old = pz_new; pz_new = tmp;
        tmp = px_old; px_old = px_new; px_new = tmp;
    }
}